// Simulation_4750233829596
// MI455X (gfx1250) — compile-verified
//
#include <hip/hip_runtime.h>

// Problem constants from the reference (T=1024, B=16384).
#define T_STEPS 1024
#define B_SIZE  16384
#define WAVES_PER_BLOCK 8

typedef __attribute__((ext_vector_type(2))) float v2f;
typedef __attribute__((ext_vector_type(4))) float v4f;
typedef __attribute__((ext_vector_type(8))) float v8f;

__device__ __forceinline__ float fast_tanh(float x) {
#if __has_builtin(__builtin_amdgcn_tanhf)
    return __builtin_amdgcn_tanhf(x);          // gfx1250 v_tanh_f32
#elif __has_builtin(__builtin_amdgcn_tanh_f32)
    return __builtin_amdgcn_tanh_f32(x);
#else
    // tanh(x) = sign(x) * (1 - e^{-2|x|}) / (1 + e^{-2|x|})
    float ax = __builtin_fabsf(x);
    float e  = __builtin_amdgcn_exp2f(ax * -2.8853900817779268f); // e^{-2|x|}
    float r  = (1.0f - e) * __builtin_amdgcn_rcpf(1.0f + e);
    return __builtin_copysignf(r, x);
#endif
}

__device__ __forceinline__ v8f wmma4(v2f a, v2f b, v8f c) {
    // V_WMMA_F32_16X16X4_F32:  D[16x16] = A[16x4] * B[4x16] + C
    // 8-arg pattern: (neg_a, A, neg_b, B, c_mod, C, reuse_a, reuse_b)
    return __builtin_amdgcn_wmma_f32_16x16x4_f32(false, a, false, b,
                                                 (short)0, c, false, false);
}

__global__ __launch_bounds__(32 * WAVES_PER_BLOCK)
void narx_tf_kernel(const float* __restrict__ inp,  // [T,B,6]
                    const float* __restrict__ tgt,  // [T,B]
                    const float* __restrict__ Wx,   // [6,6]
                    const float* __restrict__ Wu,   // [6]
                    const float* __restrict__ W1,   // [20,12]
                    const float* __restrict__ b1,   // [20]
                    const float* __restrict__ W2,   // [20]
                    const float* __restrict__ b2,   // [1]
                    float* __restrict__ out)        // [T,B]
{
    // Per-wave LDS: ni staging [16 rows x 12] and reduction scratch [16 x 20]
    __shared__ float sm_ni [WAVES_PER_BLOCK][16 * 12];
    __shared__ float sm_red[WAVES_PER_BLOCK][16 * 20];

    const int tid  = threadIdx.x;
    const int wave = tid >> 5;
    const int lane = tid & 31;
    const int half = lane >> 4;   // 0: lanes 0-15, 1: lanes 16-31
    const int row  = lane & 15;   // batch row within tile (A/B/C layouts)
    const int tile = blockIdx.x * WAVES_PER_BLOCK + wave;
    const int b0   = tile * 16;

    float* ni  = &sm_ni [wave][0];
    float* red = &sm_red[wave][0];

    // ---- one-time per-lane weight preload -------------------------------
    // db recurrence: half owns output components c0..c0+2 of its row.
    const int c0 = 3 * half;
    float wx[3][6], wu[3];
#pragma unroll
    for (int c = 0; c < 3; ++c) {
#pragma unroll
        for (int j = 0; j < 6; ++j) wx[c][j] = Wx[(c0 + c) * 6 + j];
        wu[c] = Wu[c0 + c];
    }

    // B-matrix tiles of W1^T [12 x 20], padded to N=32 (2 tiles of 16).
    // Assumed 32-bit B layout (symmetric to A): vgpr0 = K=k0+2*half,
    // vgpr1 = K=k0+2*half+1, N = lane%16 within each half.
    v2f Bt[2][3];
#pragma unroll
    for (int nt = 0; nt < 2; ++nt) {
#pragma unroll
        for (int kb = 0; kb < 3; ++kb) {
            int n = nt * 16 + row;
            int k = kb * 4 + 2 * half;
            float e0 = 0.f, e1 = 0.f;
            if (n < 20) { e0 = W1[n * 12 + k]; e1 = W1[n * 12 + k + 1]; }
            v2f bb; bb[0] = e0; bb[1] = e1;
            Bt[nt][kb] = bb;
        }
    }
    const float bias0 = b1[row];
    const float bias1 = (16 + row < 20) ? b1[16 + row] : 0.f;
    const float w2a   = W2[row];
    const float w2b   = (16 + row < 20) ? W2[16 + row] : 0.f;
    const float b2v   = b2[0];

    // ---- recurrent state: db components c0..c0+2 of row `row` ----------
    float d0 = 0.f, d1 = 0.f, d2 = 0.f;

    for (int t = 0; t < T_STEPS; ++t) {
        // x tile: 16 rows x 6 = 96 contiguous floats; lane takes 3.
        const float* xt = inp + ((size_t)t * B_SIZE + b0) * 6;
        float x0 = xt[3 * lane + 0];
        float x1 = xt[3 * lane + 1];
        float x2 = xt[3 * lane + 2];
        if (t + 1 < T_STEPS)
            __builtin_prefetch(xt + 6 * B_SIZE + 3 * lane, 0, 3);

        // teacher-forced feedback u_k = targets[t-1] (0 at t=0)
        float u = 0.f;
        if (t > 0) u = tgt[(size_t)(t - 1) * B_SIZE + b0 + row];

        // ---- db = db @ Wx^T + u @ Wu^T  (cross-half exchange + FMAs) ----
        float o0 = __shfl_xor(d0, 16, 32);
        float o1 = __shfl_xor(d1, 16, 32);
        float o2 = __shfl_xor(d2, 16, 32);
        float f0 = half ? o0 : d0, f1 = half ? o1 : d1, f2 = half ? o2 : d2;
        float f3 = half ? d0 : o0, f4 = half ? d1 : o1, f5 = half ? d2 : o2;
        float nd[3];
#pragma unroll
        for (int c = 0; c < 3; ++c) {
            float a = fmaf(wx[c][0], f0, fmaf(wx[c][1], f1, wx[c][2] * f2));
            a = fmaf(wx[c][3], f3, fmaf(wx[c][4], f4, fmaf(wx[c][5], f5, a)));
            nd[c] = fmaf(wu[c], u, a);
        }
        d0 = nd[0]; d1 = nd[1]; d2 = nd[2];

        // ---- stage ni = [db | x] into LDS (row-major [16][12]) ----------
        {
            float* p = ni + row * 12 + 3 * half;        // db part
            p[0] = d0; p[1] = d1; p[2] = d2;
            float* q = ni + (lane >> 1) * 12 + 6 + 3 * (lane & 1); // x part
            q[0] = x0; q[1] = x1; q[2] = x2;
        }

        // ---- read A fragments (2 consecutive floats per lane per K-blk) -
        const float* nr = ni + row * 12 + 2 * half;
        v2f a0 = *(const v2f*)(nr + 0);
        v2f a1 = *(const v2f*)(nr + 4);
        v2f a2 = *(const v2f*)(nr + 8);

        // ---- h_pre = ni @ W1^T + b1 via 6 chained f32 WMMAs -------------
        v8f acc0, acc1;
#pragma unroll
        for (int r = 0; r < 8; ++r) { acc0[r] = bias0; acc1[r] = bias1; }
        acc0 = wmma4(a0, Bt[0][0], acc0);
        acc1 = wmma4(a0, Bt[1][0], acc1);
        acc0 = wmma4(a1, Bt[0][1], acc0);
        acc1 = wmma4(a1, Bt[1][1], acc1);
        acc0 = wmma4(a2, Bt[0][2], acc0);
        acc1 = wmma4(a2, Bt[1][2], acc1);

        // ---- out = tanh(h_pre) @ W2^T + b2 ------------------------------
        float s[8];
#pragma unroll
        for (int r = 0; r < 8; ++r)
            s[r] = fmaf(fast_tanh(acc0[r]), w2a, fast_tanh(acc1[r]) * w2b);

        // transpose-reduce through LDS: red[m][n16] then row-sum per m
#pragma unroll
        for (int r = 0; r < 8; ++r)
            red[(r + 8 * half) * 20 + row] = s[r];

        if (lane < 16) {
            const v4f* rp = (const v4f*)(red + lane * 20);
            v4f A = rp[0], Bv = rp[1], C = rp[2], D = rp[3];
            float sum = ((A[0] + A[1]) + (A[2] + A[3]))
                      + ((Bv[0] + Bv[1]) + (Bv[2] + Bv[3]))
                      + ((C[0] + C[1]) + (C[2] + C[3]))
                      + ((D[0] + D[1]) + (D[2] + D[3]));
            out[(size_t)t * B_SIZE + b0 + lane] = sum + b2v;
        }
    }
}

extern "C" void kernel_launch(void* const* d_in, const int* in_sizes, int n_in,
                              void* d_out, int out_size, void* d_ws, size_t ws_size,
                              hipStream_t stream) {
    (void)in_sizes; (void)n_in; (void)out_size; (void)d_ws; (void)ws_size;
    const float* inp = (const float*)d_in[0];
    const float* tgt = (const float*)d_in[1];
    const float* Wx  = (const float*)d_in[2];
    const float* Wu  = (const float*)d_in[3];
    const float* W1  = (const float*)d_in[4];
    const float* b1  = (const float*)d_in[5];
    const float* W2  = (const float*)d_in[6];
    const float* b2  = (const float*)d_in[7];
    float* out = (float*)d_out;

    dim3 block(32 * WAVES_PER_BLOCK);                       // 8 waves/WG
    dim3 grid((B_SIZE / 16) / WAVES_PER_BLOCK);             // 1024 tiles / 8
    hipLaunchKernelGGL(narx_tf_kernel, grid, block, 0, stream,
                       inp, tgt, Wx, Wu, W1, b1, W2, b2, out);
}